// GCNATPConv_62723702391589
// MI455X (gfx1250) — compile-verified
//
#include <hip/hip_runtime.h>
#include <stdint.h>

typedef float v2f __attribute__((ext_vector_type(2)));
typedef float v8f __attribute__((ext_vector_type(8)));

// ---------------------------------------------------------------------------
// K0: zero workspace (agg[64N] + deg_out[N] + deg_in[N])
// ---------------------------------------------------------------------------
__global__ void k_zero(float* __restrict__ p, long long total) {
    long long i = (long long)blockIdx.x * blockDim.x + threadIdx.x;
    if (i < total) p[i] = 0.0f;
}

// ---------------------------------------------------------------------------
// K1: degree histograms. ei[0..E)=row(dst), ei[E..2E)=col(src)
// ---------------------------------------------------------------------------
__global__ void k_degree(const long long* __restrict__ ei, int E,
                         float* __restrict__ deg_out, float* __restrict__ deg_in) {
    int e = blockIdx.x * blockDim.x + threadIdx.x;
    if (e < E) {
        int r = (int)ei[e];
        int c = (int)ei[(long long)E + e];
        atomicAdd(&deg_out[r], 1.0f);
        atomicAdd(&deg_in[c], 1.0f);
    }
}

// ---------------------------------------------------------------------------
// K2: deg -> rsqrt(max(deg,1)) in place, over both arrays (2N contiguous)
// ---------------------------------------------------------------------------
__global__ void k_rsq(float* __restrict__ deg, int n2) {
    int i = blockIdx.x * blockDim.x + threadIdx.x;
    if (i < n2) deg[i] = rsqrtf(fmaxf(deg[i], 1.0f));
}

// ---------------------------------------------------------------------------
// K3: scatter  agg[row] += rsq_out[row]*rsq_in[col] * x[col]
// 16 lanes per edge, float4 per lane, 4 f32 atomics per lane.
// ---------------------------------------------------------------------------
__global__ void k_scatter(const float* __restrict__ x,
                          const long long* __restrict__ ei, int E,
                          const float* __restrict__ rsq_out,
                          const float* __restrict__ rsq_in,
                          float* __restrict__ agg) {
    long long t = (long long)blockIdx.x * blockDim.x + threadIdx.x;
    int e = (int)(t >> 4);          // edge id
    int f = ((int)t & 15) * 4;      // feature offset (0..60)
    if (e >= E) return;

    int r = (int)ei[e];
    int c = (int)ei[(long long)E + e];
    float w = rsq_out[r] * rsq_in[c];

    const float4 v = *(const float4*)(x + (size_t)c * 64 + f);
    float* a = agg + (size_t)r * 64 + f;
    atomicAdd(a + 0, w * v.x);
    atomicAdd(a + 1, w * v.y);
    atomicAdd(a + 2, w * v.z);
    atomicAdd(a + 3, w * v.w);
}

// ---------------------------------------------------------------------------
// K4: out[N,64] = agg[N,64] @ W[64,64]^T + b  via V_WMMA_F32_16X16X4_F32
// 8 waves/block: wave -> (mtile = 2*blockIdx + w/4, ntile = w%4)
// A frag (16x4 f32): lane%16 = M row, lane/16 selects K pair -> float2 load
// B frag (4x16 f32): lane%16 = N col (W row), lane/16 selects K pair
// D frag (16x16 f32): VGPR j, half h -> row m0+j+8h, col n0+lane%16
// ---------------------------------------------------------------------------
__global__ void k_gemm_wmma(const float* __restrict__ agg,
                            const float* __restrict__ W,
                            const float* __restrict__ bias,
                            float* __restrict__ out, int N) {
    const int lane = threadIdx.x & 31;
    const int wave = threadIdx.x >> 5;
    const int mt   = blockIdx.x * 2 + (wave >> 2);
    const int nt   = wave & 3;
    const int m0   = mt * 16;
    if (m0 >= N) return;                 // wave-uniform exit: EXEC stays full

    const int lm = lane & 15;            // 0..15
    const int hi = lane >> 4;            // 0 or 1
    const int n0 = nt * 16;

    int mrow = m0 + lm;
    if (mrow >= N) mrow = N - 1;         // clamp (per-lane select, EXEC untouched)

    const float* arow = agg + (size_t)mrow * 64 + 2 * hi;
    const float* wrow = W   + (size_t)(n0 + lm) * 64 + 2 * hi;

    v8f acc = {};
#pragma unroll
    for (int k = 0; k < 64; k += 4) {
        v2f a = *(const v2f*)(arow + k);
        v2f b = *(const v2f*)(wrow + k);
        // D = A(16x4) x B(4x16) + C, full f32 precision
        acc = __builtin_amdgcn_wmma_f32_16x16x4_f32(
            /*neg_a=*/false, a, /*neg_b=*/false, b,
            /*c_mod=*/(short)0, acc, /*reuse_a=*/false, /*reuse_b=*/false);
    }

    const float bb = bias[n0 + lm];
#pragma unroll
    for (int j = 0; j < 8; ++j) {
        const int m = m0 + j + 8 * hi;
        if (m < N) out[(size_t)m * 64 + n0 + lm] = acc[j] + bb;
    }
}

// ---------------------------------------------------------------------------
// launch
// ---------------------------------------------------------------------------
extern "C" void kernel_launch(void* const* d_in, const int* in_sizes, int n_in,
                              void* d_out, int out_size, void* d_ws, size_t ws_size,
                              hipStream_t stream) {
    const float*     x  = (const float*)d_in[0];
    const long long* ei = (const long long*)d_in[1];
    const float*     W  = (const float*)d_in[2];
    const float*     b  = (const float*)d_in[3];
    float*           out = (float*)d_out;

    const int N = in_sizes[0] / 64;
    const int E = in_sizes[1] / 2;

    // workspace layout (floats): agg[64N] | deg_out[N] | deg_in[N]
    float* agg     = (float*)d_ws;
    float* deg_out = agg + (size_t)N * 64;
    float* deg_in  = deg_out + N;

    // K0: zero agg + degrees
    {
        long long total = (long long)N * 66;
        int blocks = (int)((total + 255) / 256);
        k_zero<<<blocks, 256, 0, stream>>>((float*)d_ws, total);
    }
    // K1: degrees
    k_degree<<<(E + 255) / 256, 256, 0, stream>>>(ei, E, deg_out, deg_in);
    // K2: rsqrt over both degree arrays (contiguous 2N)
    k_rsq<<<(2 * N + 255) / 256, 256, 0, stream>>>(deg_out, 2 * N);
    // K3: weighted scatter-add (16 lanes per edge)
    {
        long long threads = (long long)E * 16;
        int blocks = (int)((threads + 255) / 256);
        k_scatter<<<blocks, 256, 0, stream>>>(x, ei, E, deg_out, deg_in, agg);
    }
    // K4: dense projection via f32 WMMA (+bias)
    {
        int mtiles = (N + 15) / 16;
        int blocks = (mtiles + 1) / 2;      // 2 M-tiles x 4 N-tiles per 256-thread block
        k_gemm_wmma<<<blocks, 256, 0, stream>>>(agg, W, b, out, N);
    }
}